// RobustSelfAwareSynapse42PyTorch_87935160418553
// MI455X (gfx1250) — compile-verified
//
#include <hip/hip_runtime.h>
#include <stdint.h>

// ---------------------------------------------------------------------------
// Neuromodulated synapse scan, MI455X (gfx1250).
// Memory-bound streaming scan: async global->LDS double-buffered pipeline
// (ASYNCcnt), non-temporal cache policy, float4 per thread, state in VGPRs.
// ---------------------------------------------------------------------------

typedef float v4f __attribute__((ext_vector_type(4)));

#define BLOCK 256
#define VEC 4
#define TILE (BLOCK * VEC)                     // 1024 synapses per block
#define NSTREAM 4
#define STREAM_FLOATS (BLOCK * VEC)            // 1024 floats = 4 KiB per stream/stage
#define STAGE_FLOATS (NSTREAM * STREAM_FLOATS) // 4096 floats = 16 KiB per stage

struct SynState { float w, gA, gN, gGa, gGb, tr, D, A, NE, Ca; };
struct Decays   { float ga, gb, d, ach, ne, ca; };

__device__ __forceinline__ Decays make_decays(float dt) {
  const float dt_ms = dt * 1000.0f;
  Decays k;
  k.ga  = expf(-dt_ms * (1.0f / 10.0f));    // TAU_GABA_A
  k.gb  = expf(-dt_ms * (1.0f / 150.0f));   // TAU_GABA_B
  k.d   = expf(-dt_ms * (1.0f / 200.0f));   // TAU_D
  k.ach = expf(-dt_ms * (1.0f / 150.0f));   // TAU_ACH
  k.ne  = expf(-dt_ms * (1.0f / 100.0f));   // TAU_NE
  k.ca  = expf(-dt_ms * (1.0f / 800.0f));   // TAU_CA
  return k;
}

// One reference step for one synapse; returns the synaptic current I.
__device__ __forceinline__ float syn_step(SynState& s, const Decays& k,
                                          float p, float q, float v, float r) {
  const bool spk = p > 1e-3f;
  float tr = (p > 0.0f) ? fmaf(0.1f, p, s.tr * 0.95f) : s.tr;  // TRACE_DECAY
  tr = fminf(fmaxf(tr, 0.0f), 1.0f);
  float gA  = fminf(fmaxf(s.gA + (spk ? 0.5f * p : 0.0f), 0.0f), 2.0f);
  float gN  = fminf(fmaxf(s.gN + (spk ? 0.3f * p : 0.0f), 0.0f), 1.0f);
  float gGa = fminf(fmaxf(s.gGa * k.ga + (spk ? 0.2f * p : 0.0f), 0.0f), 1.0f);
  float gGb = fminf(fmaxf(s.gGb * k.gb + (spk ? 0.1f * p : 0.0f), 0.0f), 0.5f);
  float D  = s.D * k.d + ((fabsf(r) > 0.01f) ? 0.5f * fminf(fmaxf(r, 0.0f), 2.0f) : 0.0f);
  float A  = s.A * k.ach + ((tr > 0.1f) ? 0.2f * tr : 0.0f);
  float NE = s.NE * k.ne;
  float Ca = fminf(fmaxf(fmaf(0.03f * p, q, s.Ca * k.ca), 0.0f), 2.0f);   // ALPHA_CA
  float M  = fminf(fmaxf(fmaf(0.3f, Ca, 1.0f), 0.5f), 2.0f);              // BETA_ASTRO
  float neuromod = 0.8f * (D - 1.0f) + 0.5f * (A - 1.0f) + 0.3f * (NE - 1.0f);
  float dw = 0.01f * tr * neuromod * M;                                   // BASE_LR
  float w  = fminf(fmaxf(s.w + ((tr > 0.01f) ? dw : 0.0f), 0.001f), 1.0f);
  // NMDA Mg2+ block: 1 / (1 + exp(-0.062 v) / 3.57)
  float mg = 1.0f / fmaf(__expf(-0.062f * v), (1.0f / 3.57f), 1.0f);
  float I  = w * M * ((gA + gN * mg) * (0.0f - v) + (gGa + gGb) * (-70.0f - v));
  s.w = w; s.gA = gA; s.gN = gN; s.gGa = gGa; s.gGb = gGb;
  s.tr = tr; s.D = D; s.A = A; s.NE = NE; s.Ca = Ca;
  return I;
}

// gfx1250 async DMA: copy 16B/lane from global to this lane's LDS slot.
// ASYNCcnt-tracked; NT policy keeps the 670MB stream out of the 192MB L2.
__device__ __forceinline__ void async_b128(uint32_t lds_byte,
                                           const float* base, uint32_t voff) {
  asm volatile("global_load_async_to_lds_b128 %0, %1, %2 th:TH_LOAD_NT"
               :: "v"(lds_byte), "v"(voff), "s"(base)
               : "memory");
}

__global__ __launch_bounds__(BLOCK) void synapse_pipeline_kernel(
    const float* __restrict__ pre_spike, const float* __restrict__ post_spike,
    const float* __restrict__ post_volt, const float* __restrict__ reward,
    const float* __restrict__ w_in,  const float* __restrict__ gA_in,
    const float* __restrict__ gN_in, const float* __restrict__ gGa_in,
    const float* __restrict__ gGb_in, const float* __restrict__ tr_in,
    const float* __restrict__ D_in,  const float* __restrict__ A_in,
    const float* __restrict__ NE_in, const float* __restrict__ Ca_in,
    const float* __restrict__ dt_in, float* __restrict__ out,
    int N, int T) {
  __shared__ __align__(16) float lds[2 * STAGE_FLOATS];  // 32 KiB double buffer
  const int tid = (int)threadIdx.x;
  const int n0  = (int)blockIdx.x * TILE + tid * VEC;
  if (n0 + VEC > N || T <= 0) return;  // launch config makes this uniform

  const Decays k = make_decays(dt_in[0]);

  // Initial carry -> registers (vector loads, never written back).
  v4f wv  = *(const v4f*)(w_in  + n0);
  v4f gAv = *(const v4f*)(gA_in + n0);
  v4f gNv = *(const v4f*)(gN_in + n0);
  v4f gav = *(const v4f*)(gGa_in + n0);
  v4f gbv = *(const v4f*)(gGb_in + n0);
  v4f trv = *(const v4f*)(tr_in + n0);
  v4f Dv  = *(const v4f*)(D_in  + n0);
  v4f Av  = *(const v4f*)(A_in  + n0);
  v4f NEv = *(const v4f*)(NE_in + n0);
  v4f Cav = *(const v4f*)(Ca_in + n0);
  SynState st[VEC];
#pragma unroll
  for (int l = 0; l < VEC; ++l) {
    st[l].w = wv[l];  st[l].gA = gAv[l]; st[l].gN = gNv[l];
    st[l].gGa = gav[l]; st[l].gGb = gbv[l]; st[l].tr = trv[l];
    st[l].D = Dv[l];  st[l].A = Av[l];  st[l].NE = NEv[l]; st[l].Ca = Cav[l];
  }

  // This lane's LDS slot base (addrspacecast local->flat keeps the LDS offset
  // in the low 32 bits, which is what the async instruction's VDST wants).
  const uint32_t lds_my =
      (uint32_t)(uintptr_t)(const void*)&lds[0] + (uint32_t)tid * (VEC * 4u);

  auto issue = [&](int t) {
    const uint32_t voff = ((uint32_t)t * (uint32_t)N + (uint32_t)n0) * 4u;
    const uint32_t lb = lds_my + (uint32_t)(t & 1) * (uint32_t)(STAGE_FLOATS * 4);
    async_b128(lb + 0u * (STREAM_FLOATS * 4u), pre_spike,  voff);
    async_b128(lb + 1u * (STREAM_FLOATS * 4u), post_spike, voff);
    async_b128(lb + 2u * (STREAM_FLOATS * 4u), post_volt,  voff);
    async_b128(lb + 3u * (STREAM_FLOATS * 4u), reward,     voff);
  };

  auto consume = [&](int t) {
    const int fb = (t & 1) * STAGE_FLOATS + tid * VEC;
    v4f pre  = *(const v4f*)&lds[fb + 0 * STREAM_FLOATS];
    v4f post = *(const v4f*)&lds[fb + 1 * STREAM_FLOATS];
    v4f vpo  = *(const v4f*)&lds[fb + 2 * STREAM_FLOATS];
    v4f rew  = *(const v4f*)&lds[fb + 3 * STREAM_FLOATS];
    v4f I;
#pragma unroll
    for (int l = 0; l < VEC; ++l)
      I[l] = syn_step(st[l], k, pre[l], post[l], vpo[l], rew[l]);
    __builtin_nontemporal_store(I, (v4f*)(out + (size_t)t * (size_t)N + n0));
  };

  // Software pipeline: stage t+1 in flight while computing stage t.
  issue(0);
  int t = 0;
  for (; t < T - 1; ++t) {
    issue(t + 1);
    // 4 async copies per stage; in-order completion => <=4 means stage t done.
    asm volatile("s_wait_asynccnt 4" ::: "memory");
    consume(t);
  }
  asm volatile("s_wait_asynccnt 0" ::: "memory");
  consume(t);
}

// Scalar fallback for any tail not divisible by TILE (unused for N=524288).
__global__ void synapse_tail_kernel(
    const float* __restrict__ pre_spike, const float* __restrict__ post_spike,
    const float* __restrict__ post_volt, const float* __restrict__ reward,
    const float* __restrict__ w_in,  const float* __restrict__ gA_in,
    const float* __restrict__ gN_in, const float* __restrict__ gGa_in,
    const float* __restrict__ gGb_in, const float* __restrict__ tr_in,
    const float* __restrict__ D_in,  const float* __restrict__ A_in,
    const float* __restrict__ NE_in, const float* __restrict__ Ca_in,
    const float* __restrict__ dt_in, float* __restrict__ out,
    int nStart, int N, int T) {
  const int n = nStart + (int)(blockIdx.x * blockDim.x + threadIdx.x);
  if (n >= N) return;
  const Decays k = make_decays(dt_in[0]);
  SynState s;
  s.w = w_in[n]; s.gA = gA_in[n]; s.gN = gN_in[n]; s.gGa = gGa_in[n];
  s.gGb = gGb_in[n]; s.tr = tr_in[n]; s.D = D_in[n]; s.A = A_in[n];
  s.NE = NE_in[n]; s.Ca = Ca_in[n];
  for (int t = 0; t < T; ++t) {
    const size_t idx = (size_t)t * (size_t)N + (size_t)n;
    const float I = syn_step(s, k, pre_spike[idx], post_spike[idx],
                             post_volt[idx], reward[idx]);
    __builtin_nontemporal_store(I, out + idx);
  }
}

extern "C" void kernel_launch(void* const* d_in, const int* in_sizes, int n_in,
                              void* d_out, int out_size, void* d_ws, size_t ws_size,
                              hipStream_t stream) {
  (void)n_in; (void)d_ws; (void)ws_size;
  // setup_inputs() order:
  // 0 pre_spike [T,N]   1 post_spike [T,N]  2 pre_voltage [T,N] (unused)
  // 3 post_voltage[T,N] 4 reward [T,N]
  // 5 w  6 g_ampa  7 g_nmda  8 g_gaba_a  9 g_gaba_b  10 trace
  // 11 dopamine  12 ach  13 ne  14 astro_ca  15 astro_mod (unused)  16 dt
  const float* pre  = (const float*)d_in[0];
  const float* post = (const float*)d_in[1];
  const float* pv   = (const float*)d_in[3];
  const float* rew  = (const float*)d_in[4];
  const float* w    = (const float*)d_in[5];
  const float* gA   = (const float*)d_in[6];
  const float* gN   = (const float*)d_in[7];
  const float* gGa  = (const float*)d_in[8];
  const float* gGb  = (const float*)d_in[9];
  const float* tr   = (const float*)d_in[10];
  const float* D    = (const float*)d_in[11];
  const float* A    = (const float*)d_in[12];
  const float* NE   = (const float*)d_in[13];
  const float* Ca   = (const float*)d_in[14];
  const float* dt   = (const float*)d_in[16];
  float* out = (float*)d_out;

  const int N = in_sizes[5];
  const int T = (N > 0) ? in_sizes[0] / N : 0;
  if (N <= 0 || T <= 0) return;
  (void)out_size;

  const int nVec   = (N / TILE) * TILE;
  const int blocks = nVec / TILE;
  if (blocks > 0) {
    synapse_pipeline_kernel<<<dim3(blocks), dim3(BLOCK), 0, stream>>>(
        pre, post, pv, rew, w, gA, gN, gGa, gGb, tr, D, A, NE, Ca, dt, out, N, T);
  }
  const int tail = N - nVec;
  if (tail > 0) {
    synapse_tail_kernel<<<dim3((tail + 255) / 256), dim3(256), 0, stream>>>(
        pre, post, pv, rew, w, gA, gN, gGa, gGb, tr, D, A, NE, Ca, dt, out,
        nVec, N, T);
  }
}